// IouLoss_41231686041778
// MI455X (gfx1250) — compile-verified
//
#include <hip/hip_runtime.h>
#include <cstdint>
#include <cmath>

#define TPB  256
#define NBLK 2048

// Flat pointers to __shared__ objects carry the workgroup-relative LDS offset
// in their low 32 bits (see ISA 10.2, LDS aperture).
__device__ __forceinline__ unsigned lds_off(const void* p) {
    return (unsigned)(uintptr_t)p;
}

// CDNA5 async HBM->LDS copy, GVS form: 64-bit SGPR base + signed 32-bit VGPR
// byte offset. 16B per lane, tracked by ASYNCcnt. NT hint: data is read once.
__device__ __forceinline__ void async_ld_b128_gvs(unsigned lds, unsigned off_bytes,
                                                  const void* base) {
    asm volatile("global_load_async_to_lds_b128 %0, %1, %2 th:TH_LOAD_NT"
                 :: "v"(lds), "v"(off_bytes),
                    "s"((unsigned long long)(uintptr_t)base)
                 : "memory");
}
__device__ __forceinline__ void wait_async_le2() {
    asm volatile("s_wait_asynccnt 2" ::: "memory");
}
__device__ __forceinline__ void wait_async_le0() {
    asm volatile("s_wait_asynccnt 0" ::: "memory");
}
__device__ __forceinline__ void wait_ds0() {
    asm volatile("s_wait_dscnt 0" ::: "memory");
}

__global__ __launch_bounds__(TPB) void iou_partial_kernel(
        const float4* __restrict__ A,   // boxes_pred as xywh
        const float4* __restrict__ B,   // boxes as xywh
        float* __restrict__ partials,
        long N) {                        // requires N*16 < 2^31 (holds: N = 2^23)
    __shared__ float4 sA[2][TPB];
    __shared__ float4 sB[2][TPB];

    const int      tid    = threadIdx.x;
    const long     nTiles = (N + TPB - 1) / TPB;
    const unsigned limit  = (unsigned)(N * 16L);        // first OOB byte offset
    const unsigned maxoff = limit - 16u;                // last valid element
    const unsigned stride = (unsigned)gridDim.x * TPB * 16u;

    // Per-thread LDS slot addresses (constant across the loop).
    const unsigned lA0 = lds_off(&sA[0][tid]), lA1 = lds_off(&sA[1][tid]);
    const unsigned lB0 = lds_off(&sB[0][tid]), lB1 = lds_off(&sB[1][tid]);

    float acc = 0.0f;

    auto stage = [&](int buf, unsigned off) {
        // Prior ds reads of this buffer must retire before the async engine
        // overwrites it (DS ops and async LDS writes are unordered).
        wait_ds0();
        unsigned offc = off < maxoff ? off : maxoff;   // clamp: uniform EXEC, no OOB
        async_ld_b128_gvs(buf ? lA1 : lA0, offc, A);
        async_ld_b128_gvs(buf ? lB1 : lB0, offc, B);
    };

    auto consume = [&](int buf, bool valid) {
        float4 p = buf ? sA[1][tid] : sA[0][tid];
        float4 q = buf ? sB[1][tid] : sB[0][tid];
        // xywh -> xyxy (mirror reference rounding: x1 = x + w, area from x1-x0)
        float ax1 = p.x + p.z, ay1 = p.y + p.w;
        float bx1 = q.x + q.z, by1 = q.y + q.w;
        float ltx = fmaxf(p.x, q.x), lty = fmaxf(p.y, q.y);
        float rbx = fminf(ax1, bx1), rby = fminf(ay1, by1);
        float w = fmaxf(rbx - ltx, 0.0f);
        float h = fmaxf(rby - lty, 0.0f);
        float inter = w * h;
        float areaA = (ax1 - p.x) * (ay1 - p.y);
        float areaB = (bx1 - q.x) * (by1 - q.y);
        float iou = inter / (areaA + areaB - inter);
        acc += valid ? iou : 0.0f;
    };

    long     t   = blockIdx.x;                       // uniform -> scalar control flow
    unsigned off = ((unsigned)blockIdx.x * TPB + (unsigned)tid) * 16u;
    if (t < nTiles) {
        int buf = 0;
        stage(0, off);
        for (;;) {
            long     tn   = t + (long)gridDim.x;
            bool     more = (tn < nTiles);
            unsigned offn = off + stride;
            if (more) {
                stage(buf ^ 1, offn);
                wait_async_le2();    // in-order async completion: oldest buf landed
            } else {
                wait_async_le0();
            }
            consume(buf, off < limit);
            if (!more) break;
            buf ^= 1;
            t   = tn;
            off = offn;
        }
    }

    // wave32 reduction, then cross-wave via LDS
    for (int o = 16; o > 0; o >>= 1) acc += __shfl_xor(acc, o, 32);
    __shared__ float wsum[TPB / 32];
    if ((tid & 31) == 0) wsum[tid >> 5] = acc;
    __syncthreads();
    if (tid < 32) {
        float s = (tid < TPB / 32) ? wsum[tid] : 0.0f;
        for (int o = 16; o > 0; o >>= 1) s += __shfl_xor(s, o, 32);
        if (tid == 0) partials[blockIdx.x] = s;   // every slot rewritten each launch
    }
}

__global__ __launch_bounds__(256) void iou_finalize_kernel(
        const float* __restrict__ partials, int nPart,
        float* __restrict__ out, long N) {
    __shared__ double dbuf[256];
    int tid = threadIdx.x;
    double s = 0.0;
    for (int i = tid; i < nPart; i += 256) s += (double)partials[i];
    dbuf[tid] = s;
    __syncthreads();
    if (tid < 32) {
        double t = 0.0;
        for (int j = 0; j < 8; ++j) t += dbuf[tid * 8 + j];  // all ds reads issue
        dbuf[tid] = t;                                       // before this store
    }
    __syncthreads();
    if (tid == 0) {
        double tot = 0.0;
        for (int i = 0; i < 32; ++i) tot += dbuf[i];
        double mean = tot / (double)N;
        out[0] = (float)(-log(mean));
    }
}

extern "C" void kernel_launch(void* const* d_in, const int* in_sizes, int n_in,
                              void* d_out, int out_size, void* d_ws, size_t ws_size,
                              hipStream_t stream) {
    const float4* A = (const float4*)d_in[0];   // boxes_pred, (N,4) f32
    const float4* B = (const float4*)d_in[1];   // boxes,      (N,4) f32
    long N = (long)in_sizes[0] / 4;
    float* partials = (float*)d_ws;             // NBLK floats = 8 KB scratch

    iou_partial_kernel<<<NBLK, TPB, 0, stream>>>(A, B, partials, N);
    iou_finalize_kernel<<<1, 256, 0, stream>>>(partials, NBLK, (float*)d_out, N);
}